// PatchesLoss_68624987455847
// MI455X (gfx1250) — compile-verified
//
#include <hip/hip_runtime.h>

// CDNA5 / gfx1250, wave32. Fused gaussian-smooth + gradient + reduce for
// 16x16 fp32 patches using V_WMMA_F32_16X16X4_F32 (exact fp32 matmul).

typedef float v2f __attribute__((ext_vector_type(2)));
typedef float v8f __attribute__((ext_vector_type(8)));

__device__ __forceinline__ v8f wmma4(v2f a, v2f b, v8f c) {
  // D = A(16x4) * B(4x16) + C, fp32
  return __builtin_amdgcn_wmma_f32_16x16x4_f32(false, a, false, b, (short)0, c,
                                               false, false);
}

__launch_bounds__(256)
__global__ void patch_grad_kernel(const float* __restrict__ depth,
                                  float* __restrict__ partials, int n_patches) {
  const int lane = threadIdx.x & 31;
  const int wid  = threadIdx.x >> 5;
  const int h    = lane >> 4;   // half-wave: lanes 0-15 / 16-31
  const int j    = lane & 15;   // column (B/C/D layout) or row M (A layout)

  // Separable normalized gaussian taps (sigma=2, k=3): w = [w0, w1, w0]
  const float e0 = __expf(-0.125f);
  const float w1 = 1.0f / (1.0f + 2.0f * e0);
  const float w0 = e0 * w1;

  // Band-matrix constant: value w_{K-j} masked to j<=13.
  // Serves as BOTH HwT in B-layout (P1) and Vw in A-layout (P2).
  v2f wk[4];
#pragma unroll
  for (int t = 0; t < 4; ++t) {
#pragma unroll
    for (int s = 0; s < 2; ++s) {
      const int K = 4 * t + 2 * h + s;
      const int d = K - j;
      float v = (d == 1) ? w1 : ((d == 0 || d == 2) ? w0 : 0.0f);
      if (j > 13) v = 0.0f;
      wk[t][s] = v;
    }
  }

  const int waves_per_block = blockDim.x >> 5;
  const int gwave  = blockIdx.x * waves_per_block + wid;
  const int nwaves = gridDim.x * waves_per_block;
  const int src2   = (lane & 16) | ((j + 2) & 15);  // column c+2, same half

  float acc = 0.0f;

  for (int p = gwave; p < n_patches; p += nwaves) {
    // ---- Load patch X (16x16 fp32, contiguous 1KB) in WMMA A layout:
    // lane(h,j) holds X[j][4t+2h], X[j][4t+2h+1] for t=0..3
    const float* pp = depth + (size_t)p * 256 + j * 16 + 2 * h;
    v2f a0 = *(const v2f*)(pp + 0);
    v2f a1 = *(const v2f*)(pp + 4);
    v2f a2 = *(const v2f*)(pp + 8);
    v2f a3 = *(const v2f*)(pp + 12);

    // ---- P1: U = X * HwT  (horizontal 3-tap smoothing), 4 chained WMMAs
    v8f u = {};
    u = wmma4(a0, wk[0], u);
    u = wmma4(a1, wk[1], u);
    u = wmma4(a2, wk[2], u);
    u = wmma4(a3, wk[3], u);

    // ---- Relayout U: C layout -> B layout chunks (column stays = lane;
    // only half-wave row swaps needed)
    float uxs[8];
#pragma unroll
    for (int k = 0; k < 8; ++k) uxs[k] = __shfl_xor((float)u[k], 16, 32);

    v2f b[4];
#pragma unroll
    for (int t = 0; t < 4; ++t) {
#pragma unroll
      for (int s = 0; s < 2; ++s) {
        // b[t][s] = U[4t+2h+s][j]
        const float v0 = (t < 2)  ? (float)u[4 * t + s]
                                  : uxs[4 * (t - 2) + s];        // h==0
        const float v1 = (t >= 2) ? (float)u[4 * (t - 2) + 2 + s]
                                  : uxs[4 * t + 2 + s];          // h==1
        b[t][s] = h ? v1 : v0;
      }
    }

    // ---- P2: G = Vw * U  (vertical 3-tap smoothing), 4 chained WMMAs
    // G[r][c] valid for r,c in 0..13 (14x14 smoothed field), C layout:
    // VGPR k <-> row k+8h, column j.
    v8f gm = {};
    gm = wmma4(wk[0], b[0], gm);
    gm = wmma4(wk[1], b[1], gm);
    gm = wmma4(wk[2], b[2], gm);
    gm = wmma4(wk[3], b[3], gm);

    // ---- Derivative taps + |.| + per-lane partial sum (uniform shuffles
    // first, then h-divergent accumulation; re-converges before next WMMA)
    float gs[8];
#pragma unroll
    for (int k = 0; k < 8; ++k) gs[k] = __shfl((float)gm[k], src2, 32);
    const float gh0 = __shfl_xor((float)gm[0], 16, 32);  // row 8 (for h==0)
    const float gh1 = __shfl_xor((float)gm[1], 16, 32);  // row 9 (for h==0)

    float sx = 0.0f, sy = 0.0f;
    if (h == 0) {
      // gx rows 1..7 at this column; gy rows 0..7
#pragma unroll
      for (int k = 1; k < 8; ++k) sx += fabsf((float)gm[k] - gs[k]);
#pragma unroll
      for (int k = 0; k < 6; ++k) sy += fabsf((float)gm[k] - (float)gm[k + 2]);
      sy += fabsf((float)gm[6] - gh0);
      sy += fabsf((float)gm[7] - gh1);
    } else {
      // gx rows 8..12; gy rows 8..11
#pragma unroll
      for (int k = 0; k < 5; ++k) sx += fabsf((float)gm[k] - gs[k]);
#pragma unroll
      for (int k = 0; k < 4; ++k) sy += fabsf((float)gm[k] - (float)gm[k + 2]);
    }
    if (j > 11) sx = 0.0f;            // gx columns 0..11
    if (j < 1 || j > 12) sy = 0.0f;   // gy columns (output col = j-1)
    acc += sx + sy;                   // 0.5 factor applied at the end
  }

  // ---- wave32 butterfly reduce
#pragma unroll
  for (int off = 16; off > 0; off >>= 1) acc += __shfl_xor(acc, off, 32);

  __shared__ float sm[8];
  if (lane == 0) sm[wid] = acc;
  __syncthreads();
  if (threadIdx.x == 0) {
    float s = 0.0f;
    for (int i = 0; i < (int)waves_per_block; ++i) s += sm[i];
    partials[blockIdx.x] = s;
  }
}

__launch_bounds__(256)
__global__ void finalize_kernel(const float* __restrict__ partials, float* __restrict__ out,
                                int nblocks, float scale) {
  __shared__ float sm[256];
  float s = 0.0f;
  for (int i = threadIdx.x; i < nblocks; i += 256) s += partials[i];
  sm[threadIdx.x] = s;
  __syncthreads();
#pragma unroll
  for (int off = 128; off > 0; off >>= 1) {
    if ((int)threadIdx.x < off) sm[threadIdx.x] += sm[threadIdx.x + off];
    __syncthreads();
  }
  if (threadIdx.x == 0) out[0] = sm[0] * scale;
}

extern "C" void kernel_launch(void* const* d_in, const int* in_sizes, int n_in,
                              void* d_out, int out_size, void* d_ws, size_t ws_size,
                              hipStream_t stream) {
  const float* depth = (const float*)d_in[0];
  const int n_patches = in_sizes[0] / 256;  // patch_size fixed at 16

  const int NB = 512;  // 2KB of partials in d_ws
  const int BS = 256;  // 8 wave32 per block
  float* partials = (float*)d_ws;

  patch_grad_kernel<<<NB, BS, 0, stream>>>(depth, partials, n_patches);

  // loss = 0.1 * sum_patches( mean(|gx|+|gy|) ); the taps' 0.5 factor is applied here via scale
  const float scale = 0.1f * 0.5f / 144.0f;
  finalize_kernel<<<1, 256, 0, stream>>>(partials, (float*)d_out, NB, scale);
}